// GCNEncoder_88029649698964
// MI455X (gfx1250) — compile-verified
//
#include <hip/hip_runtime.h>

// GCN forward: 2x GCNConv(64->64) + BatchNorm + ELU + global mean pool.
// N=50000 nodes, E=800000 edges, 64 features everywhere, G=64 graphs.

#define NN 50000
#define EE 800000
#define GG 64
#define BN_EPS 1e-5f

typedef __attribute__((ext_vector_type(2))) float v2f;
typedef __attribute__((ext_vector_type(8))) float v8f;

__device__ __forceinline__ void atomAddF(float* p, float v) {
  unsafeAtomicAdd(p, v);  // -> global_atomic_add_f32
}

// ---------------------------------------------------------------- zero fill
__global__ void k_zero(float* __restrict__ p, long long n) {
  long long i = (long long)blockIdx.x * blockDim.x + threadIdx.x;
  long long stride = (long long)gridDim.x * blockDim.x;
  for (; i < n; i += stride) p[i] = 0.0f;
}

// ---------------------------------------------------------------- degree
__global__ void k_degree(const int* __restrict__ dst, float* __restrict__ deg,
                         int e) {
  int i = blockIdx.x * blockDim.x + threadIdx.x;
  if (i < e) atomAddF(&deg[dst[i]], 1.0f);
}

__global__ void k_dinv(const float* __restrict__ deg, float* __restrict__ dinv,
                       int n) {
  int i = blockIdx.x * blockDim.x + threadIdx.x;
  if (i < n) dinv[i] = rsqrtf(deg[i] + 1.0f);  // self-loop included
}

// ---------------------------------------------------------------- GEMM via WMMA
// C[n,64] = A[n,64] * W[64,64], fp32, V_WMMA_F32_16X16X4_F32.
// Block = 256 threads = 8 waves. Wave w: row-tile w>>2 (2 tiles of 16 rows),
// col-tile w&3 (4 tiles of 16 cols). K looped 64 in steps of 4.
__global__ __launch_bounds__(256) void k_gemm64(const float* __restrict__ A,
                                                const float* __restrict__ W,
                                                float* __restrict__ C, int n) {
  __shared__ float lw[64 * 64];
  int tid = threadIdx.x;
  for (int i = tid * 4; i < 64 * 64; i += 256 * 4)
    *(float4*)&lw[i] = *(const float4*)&W[i];
  __syncthreads();

  int lane = tid & 31;
  int wave = tid >> 5;
  int half = lane >> 4;   // selects K pair {0,1} vs {2,3} (ISA A/B layout)
  int l16  = lane & 15;
  int r0 = blockIdx.x * 32 + (wave >> 2) * 16;
  int c0 = (wave & 3) * 16;

  int m = r0 + l16;
  const float* arow = A + (size_t)(m < n ? m : 0) * 64;  // clamp, EXEC stays full

  v8f acc = {};
  for (int k = 0; k < 64; k += 4) {
    int kk = k + half * 2;
    v2f a, b;
    a.x = arow[kk];
    a.y = arow[kk + 1];
    b.x = lw[kk * 64 + c0 + l16];
    b.y = lw[(kk + 1) * 64 + c0 + l16];
    acc = __builtin_amdgcn_wmma_f32_16x16x4_f32(false, a, false, b, (short)0,
                                                acc, false, false);
  }
  // D layout: VGPR r -> row r0 + r + half*8, col c0 + l16
  for (int r = 0; r < 8; ++r) {
    int row = r0 + r + half * 8;
    if (row < n) C[(size_t)row * 64 + c0 + l16] = acc[r];
  }
}

// ---------------------------------------------------------------- edge scatter
// msgs = h[src] * dinv[src]*dinv[dst]; atomically accumulated into out[dst].
// 16 threads per edge, float4 per thread (coalesced 256B per edge).
__global__ void k_edge(const float* __restrict__ h, const int* __restrict__ src,
                       const int* __restrict__ dst,
                       const float* __restrict__ dinv, float* __restrict__ out,
                       int e) {
  long long idx = (long long)blockIdx.x * blockDim.x + threadIdx.x;
  long long total = (long long)e * 16;
  if (idx >= total) return;
  long long ed = idx >> 4;
  int q = (int)(idx & 15);
  int s = src[ed], d = dst[ed];
  float coef = dinv[s] * dinv[d];
  float4 hv = *(const float4*)(h + (size_t)s * 64 + q * 4);
  float* op = out + (size_t)d * 64 + q * 4;
  atomAddF(op + 0, hv.x * coef);
  atomAddF(op + 1, hv.y * coef);
  atomAddF(op + 2, hv.z * coef);
  atomAddF(op + 3, hv.w * coef);
}

// ---------------------------------------------------------------- layer-1 tail
// agg += h * dinv^2 + b1; accumulate per-feature BN sum / sumsq.
__global__ __launch_bounds__(256) void k_selfloop_bn(
    float* __restrict__ agg, const float* __restrict__ h,
    const float* __restrict__ dinv, const float* __restrict__ b,
    float* __restrict__ bnsum, float* __restrict__ bnsq, int n) {
  int f = threadIdx.x & 63;
  int rib = threadIdx.x >> 6;  // 0..3 node rows per block
  float bv = b[f];
  float ls = 0.0f, lq = 0.0f;
  for (int node = blockIdx.x * 4 + rib; node < n; node += gridDim.x * 4) {
    float di = dinv[node];
    size_t o = (size_t)node * 64 + f;
    float v = agg[o] + h[o] * di * di + bv;
    agg[o] = v;
    ls += v;
    lq += v * v;
  }
  __shared__ float ss[256], sq[256];
  ss[threadIdx.x] = ls;
  sq[threadIdx.x] = lq;
  __syncthreads();
  if (rib == 0) {
    float s = ss[f] + ss[64 + f] + ss[128 + f] + ss[192 + f];
    float q = sq[f] + sq[64 + f] + sq[128 + f] + sq[192 + f];
    atomAddF(&bnsum[f], s);
    atomAddF(&bnsq[f], q);
  }
}

// ---------------------------------------------------------------- BN + ELU
__global__ void k_bn_elu(const float* __restrict__ agg, float* __restrict__ ho,
                         const float* __restrict__ bnsum,
                         const float* __restrict__ bnsq,
                         const float* __restrict__ gamma,
                         const float* __restrict__ beta, int n) {
  long long idx = (long long)blockIdx.x * blockDim.x + threadIdx.x;
  if (idx >= (long long)n * 64) return;
  int f = (int)(idx & 63);
  float invn = 1.0f / (float)n;
  float mu = bnsum[f] * invn;
  float var = bnsq[f] * invn - mu * mu;
  float v = (agg[idx] - mu) * rsqrtf(var + BN_EPS) * gamma[f] + beta[f];
  ho[idx] = v > 0.0f ? v : (__expf(v) - 1.0f);  // ELU(alpha=1)
}

// ---------------------------------------------------------------- layer-2 tail
// out += t*dinv^2 + b2; accumulate pooling sums and per-graph node counts.
__global__ void k_finalize(float* __restrict__ out, const float* __restrict__ t,
                           const float* __restrict__ dinv,
                           const float* __restrict__ b,
                           const int* __restrict__ batch,
                           float* __restrict__ psum, float* __restrict__ cnt,
                           int n) {
  long long idx = (long long)blockIdx.x * blockDim.x + threadIdx.x;
  if (idx >= (long long)n * 64) return;
  int node = (int)(idx >> 6);
  int f = (int)(idx & 63);
  float di = dinv[node];
  float v = out[idx] + t[idx] * di * di + b[f];
  out[idx] = v;
  int g = batch[node];
  atomAddF(&psum[(size_t)g * 64 + f], v);
  if (f == 0) atomAddF(&cnt[g], 1.0f);
}

__global__ void k_pool_div(float* __restrict__ rep,
                           const float* __restrict__ psum,
                           const float* __restrict__ cnt) {
  int idx = blockIdx.x * blockDim.x + threadIdx.x;
  if (idx >= GG * 64) return;
  rep[idx] = psum[idx] / fmaxf(cnt[idx >> 6], 1.0f);
}

// ================================================================ launcher
extern "C" void kernel_launch(void* const* d_in, const int* in_sizes, int n_in,
                              void* d_out, int out_size, void* d_ws,
                              size_t ws_size, hipStream_t stream) {
  const int N = NN, E = EE, G = GG;
  const float* x     = (const float*)d_in[0];
  const int*   ei    = (const int*)d_in[1];
  const int*   batch = (const int*)d_in[2];
  const float* W1    = (const float*)d_in[3];
  const float* b1    = (const float*)d_in[4];
  const float* gamma = (const float*)d_in[5];
  const float* beta  = (const float*)d_in[6];
  const float* W2    = (const float*)d_in[7];
  const float* b2    = (const float*)d_in[8];
  const int* srcv = ei;
  const int* dstv = ei + E;

  float* out = (float*)d_out;             // node outputs [N*64]
  float* rep = out + (size_t)N * 64;      // graph reps   [G*64]

  // workspace layout (floats): zero-init prefix first
  float* ws    = (float*)d_ws;
  float* deg   = ws;                        // N
  float* agg   = deg + N;                   // N*64 (also reused as t)
  float* bnsum = agg + (size_t)N * 64;      // 64
  float* bnsq  = bnsum + 64;                // 64
  float* psum  = bnsq + 64;                 // G*64
  float* cnt   = psum + (size_t)G * 64;     // G
  float* dinv  = cnt + G;                   // N   (not zeroed)
  float* h     = dinv + N;                  // N*64 (not zeroed)
  long long nzero = (long long)N + (long long)N * 64 + 64 + 64 +
                    (long long)G * 64 + G;

  // 1) zero accumulators + node-output region of d_out
  k_zero<<<2048, 256, 0, stream>>>(ws, nzero);
  k_zero<<<2048, 256, 0, stream>>>(out, (long long)N * 64);

  // 2) degrees and normalization
  k_degree<<<(E + 255) / 256, 256, 0, stream>>>(dstv, deg, E);
  k_dinv<<<(N + 255) / 256, 256, 0, stream>>>(deg, dinv, N);

  // 3) layer 1: h = x @ W1 (WMMA), scatter, self-loop+bias+BN stats, BN+ELU
  k_gemm64<<<(N + 31) / 32, 256, 0, stream>>>(x, W1, h, N);
  k_edge<<<(int)(((long long)E * 16 + 255) / 256), 256, 0, stream>>>(
      h, srcv, dstv, dinv, agg, E);
  k_selfloop_bn<<<256, 256, 0, stream>>>(agg, h, dinv, b1, bnsum, bnsq, N);
  k_bn_elu<<<(int)(((long long)N * 64 + 255) / 256), 256, 0, stream>>>(
      agg, h, bnsum, bnsq, gamma, beta, N);  // h <- ELU(BN(agg))

  // 4) layer 2: t = h @ W2 (WMMA, t reuses agg), scatter into out, finalize+pool
  k_gemm64<<<(N + 31) / 32, 256, 0, stream>>>(h, W2, agg, N);
  k_edge<<<(int)(((long long)E * 16 + 255) / 256), 256, 0, stream>>>(
      agg, srcv, dstv, dinv, out, E);
  k_finalize<<<(int)(((long long)N * 64 + 255) / 256), 256, 0, stream>>>(
      out, agg, dinv, b2, batch, psum, cnt, N);
  k_pool_div<<<(G * 64 + 255) / 256, 256, 0, stream>>>(rep, psum, cnt);
}